// CrossAtten_Layer_33689723470008
// MI455X (gfx1250) — compile-verified
//
#include <hip/hip_runtime.h>
#include <hip/hip_bf16.h>

// ---------------------------------------------------------------------------
// MI455X (gfx1250) implementation of the windowed cross-attention layer.
// Memory-bound problem (~200MB mandatory HBM traffic vs ~11 GFLOP), so the
// design goal is: coalesce the hostile (B,C,F,T) layout once, keep all
// intermediates resident in the 192MB L2 via workspace, and do every GEMM
// with native fp32 WMMA (v_wmma_f32_16x16x4_f32, K-loop of 32 per tile).
// wave32 / WGP: 256-thread blocks = 8 waves; wave h owns attention head h.
// ---------------------------------------------------------------------------

typedef __attribute__((ext_vector_type(2))) float v2f;
typedef __attribute__((ext_vector_type(8))) float v8f;

__device__ __forceinline__ v8f wmma_f32_4(v2f a, v2f b, v8f c) {
  // (neg_a, A, neg_b, B, c_mod, C, reuse_a, reuse_b)
  return __builtin_amdgcn_wmma_f32_16x16x4_f32(false, a, false, b, (short)0, c,
                                               false, false);
}

// Sizes
#define TT 500
#define BB 4
#define FF 65
#define CC 128
#define HH 8
#define ROWCF 8320          // F*C = 65*128
#define QK_DIM 260          // F*4
#define V_DIMF 1040         // F*16
#define OUT_ELEMS 16640000  // 4*128*65*500
#define KTAIL 74880         // 32*9*260
#define VTAIL 299520        // 32*9*1040

// ---------------------------------------------------------------------------
// Generic per-(b,f) 2D transpose through LDS (coalesced both directions).
// in element (r,c)  at  b*inBb + f*inBf + r*rsIn + c
// out element (c,r) at  b*outBb + f*outBf + c*rsOut + r
// ---------------------------------------------------------------------------
__global__ __launch_bounds__(256) void transpose_bf(
    const float* __restrict__ in, float* __restrict__ out, int R, int C,
    long inBb, long inBf, long rsIn, long outBb, long outBf, long rsOut,
    int tilesC) {
  __shared__ float s[32][33];
  const int b = blockIdx.z, f = blockIdx.y;
  const int tr = blockIdx.x / tilesC, tc = blockIdx.x % tilesC;
  const int r0 = tr * 32, c0 = tc * 32;
  const long ibase = (long)b * inBb + (long)f * inBf;
  const long obase = (long)b * outBb + (long)f * outBf;
  for (int i = 0; i < 4; ++i) {
    int r = r0 + threadIdx.y + i * 8;
    int c = c0 + threadIdx.x;
    if (r < R && c < C) s[threadIdx.y + i * 8][threadIdx.x] = in[ibase + (long)r * rsIn + c];
  }
  __syncthreads();
  for (int i = 0; i < 4; ++i) {
    int c = c0 + threadIdx.y + i * 8;
    int r = r0 + threadIdx.x;
    if (r < R && c < C) out[obase + (long)c * rsOut + r] = s[threadIdx.x][threadIdx.y + i * 8];
  }
}

// ---------------------------------------------------------------------------
// Fused projection + PReLU + per-head LayerNorm.
// One workgroup per (b,t). Input row (65x128) staged in LDS, padded to 80
// rows. Each wave owns 16x16 WMMA tiles; K loop = 32 x v_wmma_f32_16x16x4.
// LN stats per head accumulated with ds_add_f32 atomics while the C
// fragments stay live in VGPRs; then normalize + scatter to (bh,t,F*Dh).
// ---------------------------------------------------------------------------
__global__ __launch_bounds__(256) void proj_ln(
    const float* __restrict__ src,   // (b*T + t)*8320 + f*128 + c
    const float* __restrict__ W,     // O x 128 row-major
    const float* __restrict__ bias,  // O
    const float* __restrict__ aP,    // scalar PReLU slope
    const float* __restrict__ g, const float* __restrict__ be,  // 65*Dh
    float* __restrict__ outw,        // (b*8+h, t, 65*Dh)
    int O, int dhShift) {
  __shared__ float xs[80 * 128];
  __shared__ float sum[8], sumsq[8];
  const int bt = blockIdx.x;
  const int b = bt / TT, t = bt - b * TT;
  const int tid = threadIdx.x, lane = tid & 31, wave = tid >> 5;
  const int half = lane >> 4, l15 = lane & 15;

  if (tid < 8) { sum[tid] = 0.f; sumsq[tid] = 0.f; }
  const long base = (long)bt * ROWCF;
  for (int idx = tid; idx < 80 * 128; idx += 256)
    xs[idx] = (idx < ROWCF) ? src[base + idx] : 0.f;
  __syncthreads();

  const float a0 = aP[0];
  const int nTiles = 5 * (O >> 4);
  v8f acc[5];
  int tmA[5], tnA[5], nMine = 0;

  for (int tt = wave; tt < nTiles; tt += 8) {
    const int tm = tt % 5, tn = tt / 5;
    v8f c = {};
    const float* wrow = W + (tn * 16 + l15) * 128;
    const int arow = (tm * 16 + l15) * 128;
    for (int kk = 0; kk < 128; kk += 4) {
      const int kb = kk + half * 2;
      v2f av = *(const v2f*)(&xs[arow + kb]);   // ds_load_b64
      v2f bv = *(const v2f*)(&wrow[kb]);        // global b64, L2-hot
      c = wmma_f32_4(av, bv, c);
    }
    const float bo = bias[tn * 16 + l15];
    for (int i = 0; i < 8; ++i) {
      float v = c[i] + bo;
      c[i] = (v >= 0.f) ? v : a0 * v;
    }
    acc[nMine] = c; tmA[nMine] = tm; tnA[nMine] = tn; ++nMine;
  }

  for (int j = 0; j < nMine; ++j) {
    const int h = (tnA[j] * 16 + l15) >> dhShift;
    float ls = 0.f, lss = 0.f;
    for (int i = 0; i < 8; ++i) {
      const int m = tmA[j] * 16 + i + half * 8;
      if (m < FF) { const float v = acc[j][i]; ls += v; lss += v * v; }
    }
    atomicAdd(&sum[h], ls);
    atomicAdd(&sumsq[h], lss);
  }
  __syncthreads();

  const int Dh = 1 << dhShift;
  const int dim = FF << dhShift;
  const float inv = 1.0f / (float)dim;
  for (int j = 0; j < nMine; ++j) {
    const int o = tnA[j] * 16 + l15;
    const int h = o >> dhShift, e = o & (Dh - 1);
    const float mu = sum[h] * inv;
    const float var = sumsq[h] * inv - mu * mu;
    const float rstd = rsqrtf(var + 1e-5f);
    const long obase = ((long)(b * HH + h) * TT + t) * dim;
    for (int i = 0; i < 8; ++i) {
      const int m = tmA[j] * 16 + i + half * 8;
      if (m < FF) {
        const int col = m * Dh + e;
        outw[obase + col] = (acc[j][i] - mu) * rstd * g[col] + be[col];
      }
    }
  }
}

// ---------------------------------------------------------------------------
// Fused attention (CTX=10 window) + output projection + LN + residual.
// One workgroup per (b,t). Wave h computes head h's scores/softmax via
// shuffle reductions; all 256 threads then build the z row (65x128) in LDS
// as the p-weighted V window; WMMA does z*W_P^T; single-row LN via LDS
// atomics; residual added from the transposed x; coalesced write to yt.
// ---------------------------------------------------------------------------
__global__ __launch_bounds__(256) void atten_pproj_ln(
    const float* __restrict__ Qws, const float* __restrict__ Kws,
    const float* __restrict__ Vws, const float* __restrict__ Kbuf,
    const float* __restrict__ Vbuf, const float* __restrict__ xt,
    const float* __restrict__ WP, const float* __restrict__ bP,
    const float* __restrict__ aP, const float* __restrict__ gP,
    const float* __restrict__ beP, float* __restrict__ yt) {
  __shared__ float zs[80 * 128];
  __shared__ float pw[8][10];
  __shared__ float sA, sB;
  const int bt = blockIdx.x;
  const int b = bt / TT, t = bt - b * TT;
  const int tid = threadIdx.x, lane = tid & 31, wave = tid >> 5;
  const int half = lane >> 4, l15 = lane & 15;
  if (tid == 0) { sA = 0.f; sB = 0.f; }

  // ---- scores + softmax: wave == head ----
  {
    const int h = wave, bh = b * HH + h;
    const float* qrow = Qws + ((long)bh * TT + t) * QK_DIM;
    float sc[10];
    for (int c = 0; c < 10; ++c) sc[c] = 0.f;
    for (int d = lane; d < QK_DIM; d += 32) {
      const float qv = qrow[d];
      for (int c = 0; c < 10; ++c) {
        const int j = t + c;  // index into Kfull = [Kbuf(9) ; K(500)]
        const float* kr = (j < 9) ? (Kbuf + ((long)bh * 9 + j) * QK_DIM)
                                  : (Kws + ((long)bh * TT + (j - 9)) * QK_DIM);
        sc[c] += qv * kr[d];
      }
    }
    for (int c = 0; c < 10; ++c)
      for (int off = 16; off > 0; off >>= 1) sc[c] += __shfl_xor(sc[c], off, 32);
    const float scale = 0.0620173673f;  // 1/sqrt(260)
    float mx = -1e30f;
    for (int c = 0; c < 10; ++c) { sc[c] *= scale; mx = fmaxf(mx, sc[c]); }
    float ssum = 0.f;
    for (int c = 0; c < 10; ++c) { sc[c] = __expf(sc[c] - mx); ssum += sc[c]; }
    const float rs = 1.0f / ssum;
    if (lane < 10) pw[h][lane] = sc[lane] * rs;
  }
  __syncthreads();

  // ---- z row = p-weighted V window, into LDS ----
  for (int idx = tid; idx < 80 * 128; idx += 256) {
    float v = 0.f;
    if (idx < ROWCF) {
      const int f = idx >> 7, o = idx & 127;
      const int h = o >> 4, e = o & 15;
      const int bh = b * HH + h;
      const int col = f * 16 + e;
      for (int c = 0; c < 10; ++c) {
        const int j = t + c;
        const float* vr = (j < 9) ? (Vbuf + ((long)bh * 9 + j) * V_DIMF)
                                  : (Vws + ((long)bh * TT + (j - 9)) * V_DIMF);
        v += pw[h][c] * vr[col];
      }
    }
    zs[idx] = v;
  }
  __syncthreads();

  // ---- output projection: 40 tiles, 5 per wave ----
  const float a0 = aP[0];
  v8f acc[5];
  int tmA[5], tnA[5], nMine = 0;
  for (int tt = wave; tt < 40; tt += 8) {
    const int tm = tt % 5, tn = tt / 5;
    v8f c = {};
    const float* wrow = WP + (tn * 16 + l15) * 128;
    const int arow = (tm * 16 + l15) * 128;
    for (int kk = 0; kk < 128; kk += 4) {
      const int kb = kk + half * 2;
      v2f av = *(const v2f*)(&zs[arow + kb]);
      v2f bv = *(const v2f*)(&wrow[kb]);
      c = wmma_f32_4(av, bv, c);
    }
    const float bo = bP[tn * 16 + l15];
    for (int i = 0; i < 8; ++i) {
      float v = c[i] + bo;
      c[i] = (v >= 0.f) ? v : a0 * v;
    }
    acc[nMine] = c; tmA[nMine] = tm; tnA[nMine] = tn; ++nMine;
  }

  // ---- LN over the full 8320 row ----
  {
    float ls = 0.f, lss = 0.f;
    for (int j = 0; j < nMine; ++j)
      for (int i = 0; i < 8; ++i) {
        const int m = tmA[j] * 16 + i + half * 8;
        if (m < FF) { const float v = acc[j][i]; ls += v; lss += v * v; }
      }
    atomicAdd(&sA, ls);
    atomicAdd(&sB, lss);
  }
  __syncthreads();
  const float inv = 1.0f / (float)ROWCF;
  const float mu = sA * inv;
  const float var = sB * inv - mu * mu;
  const float rstd = rsqrtf(var + 1e-5f);
  const long obase = (long)bt * ROWCF;
  for (int j = 0; j < nMine; ++j) {
    const int o = tnA[j] * 16 + l15;
    for (int i = 0; i < 8; ++i) {
      const int m = tmA[j] * 16 + i + half * 8;
      if (m < FF) {
        const int col = m * 128 + o;
        yt[obase + col] =
            (acc[j][i] - mu) * rstd * gP[col] + beP[col] + xt[obase + col];
      }
    }
  }
}

// ---------------------------------------------------------------------------
// Buffer tails: K_buf_new / V_buf_new = last 9 time steps of K / V.
// ---------------------------------------------------------------------------
__global__ __launch_bounds__(256) void tails(const float* __restrict__ Kws,
                                             const float* __restrict__ Vws,
                                             float* __restrict__ out) {
  const int idx = blockIdx.x * 256 + threadIdx.x;
  if (idx < KTAIL) {
    const int bh = idx / (9 * QK_DIM);
    const int rem = idx - bh * 9 * QK_DIM;
    const int r = rem / QK_DIM, col = rem - r * QK_DIM;
    out[OUT_ELEMS + idx] = Kws[((long)bh * TT + 491 + r) * QK_DIM + col];
  }
  if (idx < VTAIL) {
    const int bh = idx / (9 * V_DIMF);
    const int rem = idx - bh * 9 * V_DIMF;
    const int r = rem / V_DIMF, col = rem - r * V_DIMF;
    out[OUT_ELEMS + KTAIL + idx] = Vws[((long)bh * TT + 491 + r) * V_DIMF + col];
  }
}

// ---------------------------------------------------------------------------
extern "C" void kernel_launch(void* const* d_in, const int* in_sizes, int n_in,
                              void* d_out, int out_size, void* d_ws,
                              size_t ws_size, hipStream_t stream) {
  const float* x = (const float*)d_in[0];
  const float* mc = (const float*)d_in[1];
  const float* Kbuf = (const float*)d_in[2];
  const float* Vbuf = (const float*)d_in[3];
  const float* W_Q = (const float*)d_in[4];
  const float* b_Q = (const float*)d_in[5];
  const float* a_Q = (const float*)d_in[6];
  const float* g_Q = (const float*)d_in[7];
  const float* be_Q = (const float*)d_in[8];
  const float* W_K = (const float*)d_in[9];
  const float* b_K = (const float*)d_in[10];
  const float* a_K = (const float*)d_in[11];
  const float* g_K = (const float*)d_in[12];
  const float* be_K = (const float*)d_in[13];
  const float* W_V = (const float*)d_in[14];
  const float* b_V = (const float*)d_in[15];
  const float* a_V = (const float*)d_in[16];
  const float* g_V = (const float*)d_in[17];
  const float* be_V = (const float*)d_in[18];
  const float* W_P = (const float*)d_in[19];
  const float* b_P = (const float*)d_in[20];
  const float* a_P = (const float*)d_in[21];
  const float* g_P = (const float*)d_in[22];
  const float* be_P = (const float*)d_in[23];
  float* out = (float*)d_out;

  float* ws = (float*)d_ws;
  float* xt = ws;                     // 16,640,000  (b,t,f,c)
  float* mct = xt + 16640000;         // 16,640,000
  float* Qws = mct + 16640000;        //  4,160,000  (bh,t,260)
  float* Kws = Qws + 4160000;         //  4,160,000
  float* Vws = Kws + 4160000;         // 16,640,000  (bh,t,1040)
  float* yt = Vws + 16640000;         // 16,640,000  (b,t,f,c) pre-final-transpose

  const dim3 tb(32, 8, 1);
  // (B,C,F,T) -> (B,T,F,C): per-(b,f) 128x500 transpose. tilesR=4, tilesC=16.
  transpose_bf<<<dim3(64, 65, 4), tb, 0, stream>>>(
      x, xt, 128, 500, 4160000L, 500L, 32500L, 4160000L, 128L, 8320L, 16);
  transpose_bf<<<dim3(64, 65, 4), tb, 0, stream>>>(
      mc, mct, 128, 500, 4160000L, 500L, 32500L, 4160000L, 128L, 8320L, 16);

  // Projections + PReLU + per-head LN (WMMA).
  proj_ln<<<2000, 256, 0, stream>>>(xt, W_Q, b_Q, a_Q, g_Q, be_Q, Qws, 32, 2);
  proj_ln<<<2000, 256, 0, stream>>>(mct, W_K, b_K, a_K, g_K, be_K, Kws, 32, 2);
  proj_ln<<<2000, 256, 0, stream>>>(mct, W_V, b_V, a_V, g_V, be_V, Vws, 128, 4);

  // Attention + output projection + LN + residual (WMMA).
  atten_pproj_ln<<<2000, 256, 0, stream>>>(Qws, Kws, Vws, Kbuf, Vbuf, xt, W_P,
                                           b_P, a_P, g_P, be_P, yt);

  // (B,T,F,C) -> (B,C,F,T): per-(b,f) 500x128 transpose. tilesR=16, tilesC=4.
  transpose_bf<<<dim3(64, 65, 4), tb, 0, stream>>>(
      yt, out, 500, 128, 4160000L, 128L, 8320L, 4160000L, 500L, 32500L, 4);

  // K_buf_new / V_buf_new.
  tails<<<(VTAIL + 255) / 256, 256, 0, stream>>>(Kws, Vws, out);
}